// XcodecResidualVectorQuantization_7636451852800
// MI455X (gfx1250) — compile-verified
//
#include <hip/hip_runtime.h>

typedef float v2f __attribute__((ext_vector_type(2)));
typedef float v4f __attribute__((ext_vector_type(4)));
typedef float v8f __attribute__((ext_vector_type(8)));

#define BD   8
#define DD   128
#define TT   8192
#define KK   1024
#define QQ   8
#define TILE 256          // rows (t values) per workgroup
#define PITCH 132         // float pitch: keeps 16B alignment, conflict-free frag reads
#define NTHREADS 256      // 8 waves of 32
#define ROWS_PER_WAVE 32  // 2 strips of 16
#define NTILES (KK / 16)  // 64 codebook tiles per quantizer

// butterfly combine across 16-lane halves (xor masks 1,2,4,8 stay within a half)
#define RED_STEP(MASK)                                                                 \
  {                                                                                    \
    float ov = __int_as_float(                                                         \
        __builtin_amdgcn_ds_swizzle(__float_as_int(v), ((MASK) << 10) | 0x1f));        \
    int oi = __builtin_amdgcn_ds_swizzle(ix, ((MASK) << 10) | 0x1f);                   \
    if (ov > v || (ov == v && oi < ix)) { v = ov; ix = oi; }                           \
  }

// Low 32 bits of a flat pointer to __shared__ == byte offset in the LDS aperture.
__device__ __forceinline__ unsigned lds_off32(const void* p) {
  return (unsigned)(uintptr_t)p;
}

// Async copy 32 bytes (2 x B128) per lane: global -> LDS, tracked by ASYNCcnt.
// Per ISA pseudocode the instruction offset applies to BOTH global and LDS address.
__device__ __forceinline__ void stage_async(const float* gsrc, unsigned ldsoff) {
  asm volatile(
      "global_load_async_to_lds_b128 %0, %1, off\n\t"
      "global_load_async_to_lds_b128 %0, %1, off offset:16"
      :
      : "v"(ldsoff), "v"(gsrc)
      : "memory");
}

__device__ __forceinline__ void wait_async0() {
  asm volatile("s_wait_asynccnt 0x0" ::: "memory");
}

// Precompute 0.5*||e_k||^2 for all Q*K codes into workspace.
__global__ void rvq_halfnorm_kernel(const float* __restrict__ cb, float* __restrict__ hn) {
  int k = blockIdx.x * blockDim.x + threadIdx.x;  // 0..Q*K-1
  const float* row = cb + (size_t)k * DD;
  float s = 0.0f;
  for (int d = 0; d < DD; d += 4) {
    v4f v = *(const v4f*)(row + d);
    s += v.x * v.x + v.y * v.y + v.z * v.z + v.w * v.w;
  }
  hn[k] = 0.5f * s;
}

__global__ __launch_bounds__(NTHREADS, 1) void rvq_main_kernel(
    const float* __restrict__ x,   // [B, D, T]
    const float* __restrict__ cb,  // [Q, K, D]
    const float* __restrict__ hn,  // [Q, K] half squared norms
    int* __restrict__ idx_out,     // [Q, B, T]
    float* __restrict__ quant_out  // [B, D, T]
) {
  __shared__ float res[TILE][PITCH];    // residual rows, row-major [t][d]
  __shared__ float cbt[2][16][PITCH];   // double-buffered codebook tile, [code][d]

  const int tid  = threadIdx.x;
  const int lane = tid & 31;
  const int wave = tid >> 5;
  const int m    = lane & 15;  // row / col index within 16
  const int h    = lane >> 4;  // half select
  const int b    = blockIdx.y;
  const int t0   = blockIdx.x * TILE;
  const int wrow0 = wave * ROWS_PER_WAVE;

  // per-thread staging slot: code n = tid>>4, float offset db = (tid&15)*8 (32 bytes)
  const int stage_n  = tid >> 4;
  const int stage_db = (tid & 15) * 8;
  const unsigned cbt_off[2] = {lds_off32(&cbt[0][stage_n][stage_db]),
                               lds_off32(&cbt[1][stage_n][stage_db])};

  const float* xb = x + (size_t)b * DD * TT;

  // ---- load residual tile (coalesced over t), store [t][d] ----
  for (int i = tid; i < TILE * DD; i += NTHREADS) {
    int d = i >> 8;    // i / TILE
    int tt = i & 255;  // i % TILE
    res[tt][d] = xb[(size_t)d * TT + t0 + tt];
  }
  __syncthreads();

  for (int q = 0; q < QQ; ++q) {
    const float* cbq = cb + (size_t)q * KK * DD;
    const float* hnq = hn + q * KK;

    // ---- cache A fragments for this wave's 32 rows (2 strips x 32 k-chunks) ----
    v2f a[2][32];
#pragma unroll
    for (int s = 0; s < 2; ++s)
#pragma unroll
      for (int c = 0; c < 32; ++c)
        a[s][c] = *(const v2f*)&res[wrow0 + s * 16 + m][4 * c + 2 * h];

    float bestv[2][8];
    int besti[2][8];
#pragma unroll
    for (int s = 0; s < 2; ++s)
#pragma unroll
      for (int r = 0; r < 8; ++r) { bestv[s][r] = -__builtin_huge_valf(); besti[s][r] = 0; }

    // prologue: async-stage tile 0 into buffer 0
    stage_async(cbq + (size_t)stage_n * DD + stage_db, cbt_off[0]);

#define TILE_BODY(NT, CUR, NXT)                                                        \
  {                                                                                    \
    wait_async0();       /* this wave's copies into buf CUR have landed */             \
    __syncthreads();     /* everyone's copies landed; prev compute on buf NXT done */  \
    if ((NT) + 1 < NTILES)                                                             \
      stage_async(cbq + (size_t)(((NT) + 1) * 16 + stage_n) * DD + stage_db,           \
                  cbt_off[NXT]);                                                       \
    float hnl = hnq[(NT) * 16 + m];                                                    \
    v8f acc0 = {};                                                                     \
    v8f acc1 = {};                                                                     \
    _Pragma("unroll")                                                                  \
    for (int c = 0; c < 32; ++c) {                                                     \
      v2f bf = *(const v2f*)&cbt[CUR][m][4 * c + 2 * h]; /* B[k][n]: n=m, k=4c+2h */   \
      acc0 = __builtin_amdgcn_wmma_f32_16x16x4_f32(false, a[0][c], false, bf,          \
                                                   (short)0, acc0, false, false);      \
      acc1 = __builtin_amdgcn_wmma_f32_16x16x4_f32(false, a[1][c], false, bf,          \
                                                   (short)0, acc1, false, false);      \
    }                                                                                  \
    int kbase = (NT) * 16 + m;                                                         \
    _Pragma("unroll")                                                                  \
    for (int r = 0; r < 8; ++r) {                                                      \
      float s0 = acc0[r] - hnl; /* strict > keeps earliest k within a lane */          \
      if (s0 > bestv[0][r]) { bestv[0][r] = s0; besti[0][r] = kbase; }                 \
      float s1 = acc1[r] - hnl;                                                        \
      if (s1 > bestv[1][r]) { bestv[1][r] = s1; besti[1][r] = kbase; }                 \
    }                                                                                  \
  }

    // ---- sweep all K codes, double-buffered: compute buf k&1, async-fill 1-(k&1) ----
    for (int nt = 0; nt < NTILES; nt += 2) {
      TILE_BODY(nt, 0, 1)
      TILE_BODY(nt + 1, 1, 0)
    }
#undef TILE_BODY

    // ---- cross-lane argmax within each 16-lane half (tie -> smaller k) ----
#pragma unroll
    for (int s = 0; s < 2; ++s)
#pragma unroll
      for (int r = 0; r < 8; ++r) {
        float v = bestv[s][r];
        int ix = besti[s][r];
        RED_STEP(1)
        RED_STEP(2)
        RED_STEP(4)
        RED_STEP(8)
        bestv[s][r] = v;
        besti[s][r] = ix;  // now uniform across the half
      }

    // ---- write indices (one writer per half) ----
    if (m == 0) {
#pragma unroll
      for (int s = 0; s < 2; ++s)
#pragma unroll
        for (int r = 0; r < 8; ++r) {
          int row = wrow0 + s * 16 + h * 8 + r;  // C-tile M = r + 8h, strip offset 16s
          idx_out[((size_t)q * BD + b) * TT + t0 + row] = besti[s][r];
        }
    }

    // ---- residual update: 16 lanes of a half update that half's 16 rows ----
#pragma unroll
    for (int s = 0; s < 2; ++s)
#pragma unroll
      for (int r = 0; r < 8; ++r) {
        int row = wrow0 + s * 16 + h * 8 + r;
        int kw = besti[s][r];
        const float* e = cbq + (size_t)kw * DD + m * 8;
        v4f e0 = *(const v4f*)e;
        v4f e1 = *(const v4f*)(e + 4);
        float* rp = &res[row][m * 8];
        v4f r0 = *(v4f*)rp;
        v4f r1 = *(v4f*)(rp + 4);
        r0 -= e0;
        r1 -= e1;
        *(v4f*)rp = r0;
        *(v4f*)(rp + 4) = r1;
      }
  }  // q loop

  // ---- epilogue: sum of quants == x - final residual ----
  __syncthreads();
  float* outb = quant_out + (size_t)b * DD * TT;
  for (int i = tid; i < TILE * DD; i += NTHREADS) {
    int d = i >> 8;
    int tt = i & 255;
    size_t g = (size_t)d * TT + t0 + tt;
    outb[g] = xb[g] - res[tt][d];
  }
}

extern "C" void kernel_launch(void* const* d_in, const int* in_sizes, int n_in,
                              void* d_out, int out_size, void* d_ws, size_t ws_size,
                              hipStream_t stream) {
  const float* x = (const float*)d_in[0];   // embeddings [8,128,8192]
  const float* cb = (const float*)d_in[1];  // codebooks  [8,1024,128]
  float* hn = (float*)d_ws;                 // 8192 floats of scratch

  int* idx_out = (int*)d_out;                               // [Q,B,T] int32
  float* quant_out = (float*)d_out + (size_t)QQ * BD * TT;  // [B,D,T] f32

  rvq_halfnorm_kernel<<<(QQ * KK) / 256, 256, 0, stream>>>(cb, hn);

  dim3 grid(TT / TILE, BD);
  rvq_main_kernel<<<grid, NTHREADS, 0, stream>>>(x, cb, hn, idx_out, quant_out);
}